// lstm_model_89283780149394
// MI455X (gfx1250) — compile-verified
//
#include <hip/hip_runtime.h>
#include <hip/hip_bf16.h>
#include <math.h>

typedef __attribute__((ext_vector_type(2))) float v2f;
typedef __attribute__((ext_vector_type(8))) float v8f;

#define HID   512
#define BATCH 128
#define SEQL  512
#define GATES (4 * HID)   // 2048

__device__ __forceinline__ float sigmoidf(float x) {
    return 1.0f / (1.0f + __expf(-x));
}

// Zero-init h0 and c.
__global__ void lstm_zero_kernel(float* __restrict__ h0, float* __restrict__ c, int n) {
    int i = blockIdx.x * blockDim.x + threadIdx.x;
    if (i < n) { h0[i] = 0.0f; c[i] = 0.0f; }
}

// One LSTM timestep, fully fused:
//   gates = emb[input[t]] @ W_ih + h_in @ W_hh + b_lstm
//   c = sig(f)*c + sig(i)*tanh(g);  h_out = sig(o)*tanh(c)
// Block: 128 threads = 4 wave32; wave w computes TWO 16x16 M-tiles of gate w
// (32 batch rows x 16 gate cols), sharing B fragments across the two tiles
// -> 4 WMMAs per 8 loads in the K-loop.
// Grid: (HID/16, BATCH/32) = (32, 4).
__global__ __launch_bounds__(128) void lstm_step_kernel(
        const int* __restrict__ input, const float* __restrict__ emb,
        const float* __restrict__ W_ih, const float* __restrict__ W_hh,
        const float* __restrict__ b_lstm,
        const float* __restrict__ h_in, float* __restrict__ h_out,
        float* __restrict__ c, int t) {
    const int wave = threadIdx.x >> 5;   // gate id: 0=i 1=f 2=g 3=o
    const int lane = threadIdx.x & 31;
    const int lh   = lane & 15;          // lane within half-wave
    const int hi   = lane >> 4;          // half-wave select
    const int bt   = blockIdx.y;         // batch super-tile (32 rows)
    const int ht   = blockIdx.x;         // hidden tile (16 cols)

    const int m0  = bt * 32 + lh;                // batch row, tile 0
    const int m1  = m0 + 16;                     // batch row, tile 1
    const int col = wave * HID + ht * 16 + lh;   // gate-matrix column (B operand)

    // A (16x4 f32): lanes 0-15 hold K={0,1}, lanes 16-31 hold K={2,3} -> +2*hi.
    const int e0 = input[t * BATCH + m0];
    const int e1 = input[t * BATCH + m1];
    const float* aX0 = emb  + (long)e0 * HID + 2 * hi;
    const float* aX1 = emb  + (long)e1 * HID + 2 * hi;
    const float* aH0 = h_in + (long)m0 * HID + 2 * hi;
    const float* aH1 = h_in + (long)m1 * HID + 2 * hi;
    // B (4x16 f32): same K striping across half-waves, column = col.
    const float* bX = W_ih + (long)(2 * hi) * GATES + col;
    const float* bH = W_hh + (long)(2 * hi) * GATES + col;

    v8f acc0 = {};
    v8f acc1 = {};
#pragma unroll 4
    for (int k = 0; k < HID; k += 4) {
        v2f ax0 = *(const v2f*)(aX0 + k);
        v2f ax1 = *(const v2f*)(aX1 + k);
        v2f ah0 = *(const v2f*)(aH0 + k);
        v2f ah1 = *(const v2f*)(aH1 + k);
        const long r0 = (long)k * GATES;
        v2f bx, bh;
        bx.x = bX[r0];  bx.y = bX[r0 + GATES];
        bh.x = bH[r0];  bh.y = bH[r0 + GATES];
        // D = A*B + C, f32 16x16x4; B fragments shared by both M-tiles.
        acc0 = __builtin_amdgcn_wmma_f32_16x16x4_f32(false, ax0, false, bx,
                                                     (short)0, acc0, false, false);
        acc1 = __builtin_amdgcn_wmma_f32_16x16x4_f32(false, ax1, false, bx,
                                                     (short)0, acc1, false, false);
        acc0 = __builtin_amdgcn_wmma_f32_16x16x4_f32(false, ah0, false, bh,
                                                     (short)0, acc0, false, false);
        acc1 = __builtin_amdgcn_wmma_f32_16x16x4_f32(false, ah1, false, bh,
                                                     (short)0, acc1, false, false);
    }

    // C/D layout: lane(0-15) vgpr r -> (M=r, N=lane); lane(16-31) -> (M=r+8, N=lane-16)
    __shared__ float g4[4][32][16];
    const float bias = b_lstm[col];
#pragma unroll
    for (int r = 0; r < 8; ++r) {
        g4[wave][r + 8 * hi][lh]      = acc0[r] + bias;
        g4[wave][16 + r + 8 * hi][lh] = acc1[r] + bias;
    }
    __syncthreads();

    // Elementwise LSTM cell update: 512 elements over 128 threads.
#pragma unroll
    for (int e = threadIdx.x; e < 512; e += 128) {
        const int mm = e >> 4;           // 0..31 (row within super-tile)
        const int nn = e & 15;
        const float ig = sigmoidf(g4[0][mm][nn]);
        const float fg = sigmoidf(g4[1][mm][nn]);
        const float gg = tanhf(g4[2][mm][nn]);
        const float og = sigmoidf(g4[3][mm][nn]);
        const int b = bt * 32 + mm, j = ht * 16 + nn;
        const float cc = fg * c[b * HID + j] + ig * gg;
        c[b * HID + j]     = cc;
        h_out[b * HID + j] = og * tanhf(cc);
    }
}

// decode = relu(h @ W_dec + b_dec) [128x100]; out = decode @ W_fc + b_fc [128x2]
__global__ __launch_bounds__(128) void lstm_head_kernel(
        const float* __restrict__ h, const float* __restrict__ W_dec,
        const float* __restrict__ b_dec, const float* __restrict__ W_fc,
        const float* __restrict__ b_fc, float* __restrict__ out) {
    __shared__ float dec[100];
    const int b = blockIdx.x;
    if (threadIdx.x < 100) {
        float s = b_dec[threadIdx.x];
        const float* hr = h + b * HID;
        for (int k = 0; k < HID; ++k)
            s = fmaf(hr[k], W_dec[k * 100 + threadIdx.x], s);
        dec[threadIdx.x] = fmaxf(s, 0.0f);
    }
    __syncthreads();
    if (threadIdx.x < 2) {
        float s = b_fc[threadIdx.x];
        for (int k = 0; k < 100; ++k)
            s = fmaf(dec[k], W_fc[k * 2 + threadIdx.x], s);
        out[b * 2 + threadIdx.x] = s;
    }
}

extern "C" void kernel_launch(void* const* d_in, const int* in_sizes, int n_in,
                              void* d_out, int out_size, void* d_ws, size_t ws_size,
                              hipStream_t stream) {
    const int*   input  = (const int*)  d_in[0];
    const float* emb    = (const float*)d_in[1];
    const float* W_ih   = (const float*)d_in[2];
    const float* W_hh   = (const float*)d_in[3];
    const float* b_lstm = (const float*)d_in[4];
    const float* W_dec  = (const float*)d_in[5];
    const float* b_dec  = (const float*)d_in[6];
    const float* W_fc   = (const float*)d_in[7];
    const float* b_fc   = (const float*)d_in[8];
    float* out = (float*)d_out;

    // Workspace: h ping-pong + c (3 * 128*512 floats = 768 KB).
    float* h0 = (float*)d_ws;
    float* h1 = h0 + BATCH * HID;
    float* c  = h1 + BATCH * HID;

    lstm_zero_kernel<<<(BATCH * HID + 255) / 256, 256, 0, stream>>>(h0, c, BATCH * HID);

    dim3 grid(HID / 16, BATCH / 32);  // 32 x 4 workgroups, 4 waves each
    for (int t = 0; t < SEQL; ++t) {
        float* hin  = (t & 1) ? h1 : h0;
        float* hout = (t & 1) ? h0 : h1;
        lstm_step_kernel<<<grid, 128, 0, stream>>>(input, emb, W_ih, W_hh, b_lstm,
                                                   hin, hout, c, t);
    }
    // SEQL is even -> final h lives in h0.
    lstm_head_kernel<<<BATCH, 128, 0, stream>>>(h0, W_dec, b_dec, W_fc, b_fc, out);
}